// Attention_19653770346936
// MI455X (gfx1250) — compile-verified
//
#include <hip/hip_runtime.h>
#include <hip/hip_bf16.h>
#include <math.h>

typedef float v2f __attribute__((ext_vector_type(2)));
typedef float v8f __attribute__((ext_vector_type(8)));
typedef unsigned int v4u __attribute__((ext_vector_type(4)));
typedef int v8i __attribute__((ext_vector_type(8)));
typedef int v4i __attribute__((ext_vector_type(4)));

// Problem constants (from reference)
constexpr int L   = 50;     // window tokens + cls
constexpr int Lp  = 64;     // padded rows (4 x 16 tiles)
constexpr int Dm  = 768;    // model dim
constexpr int Hh  = 12;     // heads
constexpr int Sd  = 64;     // head size
constexpr int Bw  = 2048;   // batch*windows
// TDM pads LDS by 4 DWORDs every 256 DWORDs -> row stride 768+12 = 780 floats
// (780 mod 64 banks = 12 -> conflict-free A-tile column access).
constexpr int XST = 780;
constexpr int QST = 68;     // LDS row stride for 64-wide tiles

// Column mapping inside a TDM-padded 768-wide row: k -> k + 4*(k/256).
__device__ __forceinline__ int wcol(int k) { return k + ((k >> 8) << 2); }

// ---- Tensor Data Mover: 2-D fp32 tile (cols x rows) global -> LDS ----------
// D# layout per CDNA5 ISA. pad: 4 DW per 256 DW.
__device__ __forceinline__ void tdm_load_2d_f32(float* lds_base,
                                                const float* gsrc,
                                                int rows, int cols) {
  unsigned long long ga = (unsigned long long)(uintptr_t)gsrc;
  unsigned int ldso = (unsigned int)(uintptr_t)lds_base;  // LDS byte offset
  v4u g0;
  g0[0] = 1u;                                             // count=1 (valid)
  g0[1] = ldso;                                           // lds_addr
  g0[2] = (unsigned int)(ga & 0xffffffffull);             // global_addr lo
  g0[3] = (unsigned int)((ga >> 32) & 0x1ffffffull) | (2u << 30); // hi | type=2
  v8i g1;
  g1[0] = (2 << 16)        // data_size = 4 bytes
        | (1 << 20)        // pad_enable
        | (7 << 22)        // pad_interval code 7 = 256 DWORDs
        | (3 << 25);       // pad_amount  code 3 = 4 DWORDs
  g1[1] = (cols & 0xffff) << 16;                       // tensor_dim0[15:0]
  g1[2] = (cols >> 16) | ((rows & 0xffff) << 16);      // dim0 hi | dim1 lo
  g1[3] = (rows >> 16) | ((cols & 0xffff) << 16);      // dim1 hi | tile_dim0
  g1[4] = (rows & 0xffff);                             // tile_dim1 (dim2=0)
  g1[5] = cols;                                        // tensor_dim0_stride lo
  g1[6] = 0;
  g1[7] = 0;
  v4i gz4 = {};
  v8i gz8 = {};
  // 6-arg toolchain signature: (v4u, v8i, v4i, v4i, v8i, cpol)
  __builtin_amdgcn_tensor_load_to_lds(g0, g1, gz4, gz4, gz8, 0);
}

// ---- WMMA f32 16x16x4 wrappers ----------------------------------------------
// A (16x4 f32): VGPR0: lanes0-15 -> A[m][k0], lanes16-31 -> A[m][k0+2]
//               VGPR1: lanes0-15 -> A[m][k0+1], lanes16-31 -> A[m][k0+3]
// -> per lane a contiguous (k, k+1) pair: one 8-byte ds load.
__device__ __forceinline__ v2f tileA_rowmajor(const float* base, int row_base,
                                              int stride, int kb) {
  int l   = threadIdx.x & 31;
  int row = row_base + (l & 15);
  int k   = kb + ((l >> 4) << 1);
  return *(const v2f*)(base + row * stride + k);
}

// A-tile load from a TDM-padded 768-wide LDS buffer.
__device__ __forceinline__ v2f tileA_padded(const float* base, int row_base,
                                            int kb) {
  int l   = threadIdx.x & 31;
  int row = row_base + (l & 15);
  int k   = kb + ((l >> 4) << 1);
  return *(const v2f*)(base + row * XST + wcol(k));
}

// B (4x16) from a row-major K x N matrix: B[k][n] = M[k][n]; per lane the
// (k, k+1) pair is `stride` apart -> two scalar loads.
__device__ __forceinline__ v2f tileB_rowmajor(const float* base, int stride,
                                              int kb, int ncol) {
  int l = threadIdx.x & 31;
  int k = kb + ((l >> 4) << 1);
  const float* p = base + k * stride + ncol + (l & 15);
  v2f r; r.x = p[0]; r.y = p[stride];
  return r;
}

__device__ __forceinline__ v8f wmma4(v2f a, v2f b, v8f c) {
  return __builtin_amdgcn_wmma_f32_16x16x4_f32(false, a, false, b,
                                               (short)0, c, false, false);
}

// Stage a [L x Dm] fp32 panel into a padded [Lp x XST] LDS buffer:
// wave 0 drives the TDM, waves 1..7 zero the pad rows.
__device__ __forceinline__ void stage_panel(float* lds, const float* gsrc) {
  const int tid = threadIdx.x;
  if (tid < 32) {
    tdm_load_2d_f32(lds, gsrc, L, Dm);
    __builtin_amdgcn_s_wait_tensorcnt(0);
  } else {
    for (int i = tid - 32; i < (Lp - L) * XST; i += 224)
      lds[L * XST + i] = 0.f;
  }
  __syncthreads();
}

// ---- Kernel 1: fused QKV projection + biased attention + softmax + A*V ------
__global__ __launch_bounds__(256) void attn_fused_kernel(
    const float* __restrict__ x,  const float* __restrict__ wq,
    const float* __restrict__ bq, const float* __restrict__ wk,
    const float* __restrict__ wv, const float* __restrict__ bv,
    const float* __restrict__ pb, const int*   __restrict__ idx,
    float* __restrict__ a_out,    float* __restrict__ ctx_ws) {
  __shared__ float xs [Lp * XST];   // x[b] staged once by TDM, reused 36x
  __shared__ float ksh[Lp * QST];   // k head slice
  __shared__ float vsh[Lp * QST];   // v head slice
  __shared__ float qat[Lp * QST];   // q, then attention probs (reuse)

  const int b   = blockIdx.x;
  const int tid = threadIdx.x;

  stage_panel(xs, x + (size_t)b * L * Dm);

  const int w    = tid >> 5;       // wave id 0..7
  const int l    = tid & 31;
  const int mt   = w & 3;          // output row tile 0..3
  const int nt0  = (w >> 2) * 2;   // output col tiles {0,1} or {2,3}
  const int nt1  = nt0 + 1;
  const int hi   = l >> 4;         // lane half
  const int ncol = l & 15;

  for (int h = 0; h < Hh; ++h) {
    // ---- q/k/v head projections: (64x768) x (768x64), K-loop of 192 WMMAs --
#define PROJ_MAT(Wm, DST, HASB, BIASPTR)                                    \
    do {                                                                    \
      v8f acc0 = {}; v8f acc1 = {};                                         \
      const int c0 = h * Sd + nt0 * 16, c1 = h * Sd + nt1 * 16;             \
      for (int kb = 0; kb < Dm; kb += 4) {                                  \
        v2f a  = tileA_padded(xs, mt * 16, kb);                             \
        v2f b0 = tileB_rowmajor(Wm, Dm, kb, c0);                            \
        v2f b1 = tileB_rowmajor(Wm, Dm, kb, c1);                            \
        acc0 = wmma4(a, b0, acc0);                                          \
        acc1 = wmma4(a, b1, acc1);                                          \
      }                                                                     \
      float bias0 = (HASB) ? (BIASPTR)[c0 + ncol] : 0.f;                    \
      float bias1 = (HASB) ? (BIASPTR)[c1 + ncol] : 0.f;                    \
      for (int r = 0; r < 8; ++r) {                                         \
        int m = mt * 16 + r + hi * 8;                                       \
        DST[m * QST + nt0 * 16 + ncol] = acc0[r] + bias0;                   \
        DST[m * QST + nt1 * 16 + ncol] = acc1[r] + bias1;                   \
      }                                                                     \
    } while (0)

    PROJ_MAT(wq, qat, 1, bq);
    PROJ_MAT(wk, ksh, 0, bq);
    PROJ_MAT(wv, vsh, 1, bv);
#undef PROJ_MAT
    __syncthreads();

    // ---- scores = q . k^T  (k^T tile == A-style load from ksh) -------------
    v8f s0 = {}; v8f s1 = {};
    for (int kb = 0; kb < Sd; kb += 4) {
      v2f a  = tileA_rowmajor(qat, mt * 16, QST, kb);
      v2f b0 = tileA_rowmajor(ksh, nt0 * 16, QST, kb);
      v2f b1 = tileA_rowmajor(ksh, nt1 * 16, QST, kb);
      s0 = wmma4(a, b0, s0);
      s1 = wmma4(a, b1, s1);
    }
    __syncthreads();   // everyone done reading q before qat becomes attn

    // scale + relative-position bias gather, masked write into qat
    {
      const float scale = 0.125f;  // 1/sqrt(64)
      for (int r = 0; r < 8; ++r) {
        int m  = mt * 16 + r + hi * 8;
        int n0 = nt0 * 16 + ncol, n1 = nt1 * 16 + ncol;
        float v0 = -1e30f, v1 = -1e30f;
        if (m < L && n0 < L) v0 = s0[r] * scale + pb[idx[m * L + n0] * Hh + h];
        if (m < L && n1 < L) v1 = s1[r] * scale + pb[idx[m * L + n1] * Hh + h];
        qat[m * QST + n0] = v0;
        qat[m * QST + n1] = v1;
      }
    }
    __syncthreads();

    // ---- row softmax (L=50 is tiny: one thread per row), emit `a` ----------
    if (tid < Lp) {
      int m = tid;
      float* row = &qat[m * QST];
      if (m >= L) {
        for (int n = 0; n < Lp; ++n) row[n] = 0.f;   // keep A*V tiles clean
      } else {
        float mx = -1e30f;
        for (int n = 0; n < L; ++n) mx = fmaxf(mx, row[n]);
        float sum = 0.f;
        for (int n = 0; n < L; ++n) { float e = __expf(row[n] - mx); row[n] = e; sum += e; }
        float inv = 1.f / sum;
        float* ag = a_out + (((size_t)b * Hh + h) * L + m) * L;
        for (int n = 0; n < L; ++n) { float p = row[n] * inv; row[n] = p; ag[n] = p; }
        for (int n = L; n < Lp; ++n) row[n] = 0.f;
      }
    }
    __syncthreads();

    // ---- ctx = attn . v  -> workspace [B, L, D] head-concatenated ----------
    v8f c0 = {}; v8f c1 = {};
    for (int kb = 0; kb < Sd; kb += 4) {
      v2f a  = tileA_rowmajor(qat, mt * 16, QST, kb);
      v2f b0 = tileB_rowmajor(vsh, QST, kb, nt0 * 16);
      v2f b1 = tileB_rowmajor(vsh, QST, kb, nt1 * 16);
      c0 = wmma4(a, b0, c0);
      c1 = wmma4(a, b1, c1);
    }
    {
      float* cw = ctx_ws + (size_t)b * L * Dm + h * Sd;
      for (int r = 0; r < 8; ++r) {
        int m = mt * 16 + r + hi * 8;
        if (m < L) {
          cw[(size_t)m * Dm + nt0 * 16 + ncol] = c0[r];
          cw[(size_t)m * Dm + nt1 * 16 + ncol] = c1[r];
        }
      }
    }
    __syncthreads();   // qat/ksh/vsh recycled next head
  }
}

// ---- Kernel 2: out = ctx @ wo + bo  ----------------------------------------
__global__ __launch_bounds__(256) void out_proj_kernel(
    const float* __restrict__ ctx, const float* __restrict__ wo,
    const float* __restrict__ bo,  float* __restrict__ out) {
  __shared__ float cs[Lp * XST];
  const int b = blockIdx.x, tid = threadIdx.x;

  stage_panel(cs, ctx + (size_t)b * L * Dm);

  const int w = tid >> 5, l = tid & 31;
  const int mt = w & 3, half = w >> 2;
  const int hi = l >> 4, ncol = l & 15;
  float* og = out + (size_t)b * L * Dm;

  for (int t = 0; t < 12; ++t) {               // 48 col tiles over 8 waves
    int nt0 = half * 24 + 2 * t, nt1 = nt0 + 1;
    v8f a0 = {}, a1 = {};
    for (int kb = 0; kb < Dm; kb += 4) {
      v2f a  = tileA_padded(cs, mt * 16, kb);
      v2f b0 = tileB_rowmajor(wo, Dm, kb, nt0 * 16);
      v2f b1 = tileB_rowmajor(wo, Dm, kb, nt1 * 16);
      a0 = wmma4(a, b0, a0);
      a1 = wmma4(a, b1, a1);
    }
    float bias0 = bo[nt0 * 16 + ncol], bias1 = bo[nt1 * 16 + ncol];
    for (int r = 0; r < 8; ++r) {
      int m = mt * 16 + r + hi * 8;
      if (m < L) {
        og[(size_t)m * Dm + nt0 * 16 + ncol] = a0[r] + bias0;
        og[(size_t)m * Dm + nt1 * 16 + ncol] = a1[r] + bias1;
      }
    }
  }
}

extern "C" void kernel_launch(void* const* d_in, const int* in_sizes, int n_in,
                              void* d_out, int out_size, void* d_ws, size_t ws_size,
                              hipStream_t stream) {
  const float* x  = (const float*)d_in[0];
  const float* wq = (const float*)d_in[1];
  const float* bq = (const float*)d_in[2];
  const float* wk = (const float*)d_in[3];
  const float* wv = (const float*)d_in[4];
  const float* bv = (const float*)d_in[5];
  const float* wo = (const float*)d_in[6];
  const float* bo = (const float*)d_in[7];
  const float* pb = (const float*)d_in[8];
  const int*   idx = (const int*)d_in[9];

  float* out   = (float*)d_out;                       // [B, L, D]
  float* a_out = out + (size_t)Bw * L * Dm;           // [B, H, L, L]
  float* ctx   = (float*)d_ws;                        // [B, L, D] scratch

  attn_fused_kernel<<<Bw, 256, 0, stream>>>(x, wq, bq, wk, wv, bv, pb, idx,
                                            a_out, ctx);
  out_proj_kernel<<<Bw, 256, 0, stream>>>(ctx, wo, bo, out);
}